// InformationPrototype_82360292868467
// MI455X (gfx1250) — compile-verified
//
#include <hip/hip_runtime.h>
#include <hip/hip_bf16.h>
#include <math.h>

typedef __attribute__((ext_vector_type(2))) float v2f;
typedef __attribute__((ext_vector_type(8))) float v8f;

#define N_ROWS 8192
#define DIM    512
#define NCLS   100

// ---------- Kernel 1: row L2 norms -> xn = x / max(||x||, eps) ----------
__global__ void knorm(const float* __restrict__ x, float* __restrict__ xn) {
    int row  = blockIdx.x;
    int lane = threadIdx.x;                       // wave32
    const float4* xr = (const float4*)(x + (size_t)row * DIM);
    float4 v[4];
    float ss = 0.f;
#pragma unroll
    for (int i = 0; i < 4; ++i) {
        v[i] = xr[lane + 32 * i];                 // coalesced 512B / iter
        ss += v[i].x * v[i].x + v[i].y * v[i].y + v[i].z * v[i].z + v[i].w * v[i].w;
    }
#pragma unroll
    for (int off = 16; off >= 1; off >>= 1) ss += __shfl_xor(ss, off, 32);
    float s = 1.0f / fmaxf(sqrtf(ss), 1e-8f);
    float4* xo = (float4*)(xn + (size_t)row * DIM);
#pragma unroll
    for (int i = 0; i < 4; ++i) {
        float4 o; o.x = v[i].x * s; o.y = v[i].y * s; o.z = v[i].z * s; o.w = v[i].w * s;
        xo[lane + 32 * i] = o;
    }
}

// ---------- Kernel 2: softmax max-prob, argmax, class counts ----------
__global__ void ksoft(const float* __restrict__ logits, float* __restrict__ maxprob,
                      int* __restrict__ cls, float* __restrict__ counts) {
    int row  = blockIdx.x;
    int lane = threadIdx.x;
    const float* lr = logits + (size_t)row * NCLS;
    float mv = -INFINITY; int mi = 0x7fffffff;
    for (int j = lane; j < NCLS; j += 32) {
        float v = lr[j];
        if (v > mv || (v == mv && j < mi)) { mv = v; mi = j; }
    }
#pragma unroll
    for (int off = 16; off >= 1; off >>= 1) {
        float ov = __shfl_xor(mv, off, 32);
        int   oi = __shfl_xor(mi, off, 32);
        if (ov > mv || (ov == mv && oi < mi)) { mv = ov; mi = oi; }
    }
    float se = 0.f;
    for (int j = lane; j < NCLS; j += 32) se += __expf(lr[j] - mv);
#pragma unroll
    for (int off = 16; off >= 1; off >>= 1) se += __shfl_xor(se, off, 32);
    if (lane == 0) {
        maxprob[row] = 1.0f / se;                 // softmax value at argmax
        cls[row]     = mi;
        atomicAdd(&counts[mi], 1.0f);
    }
}

// ---------- Kernel 3: G = xn^T @ x (512x512), WMMA f32, 2x4 register blocking ----
// Each wave computes a 32x64 block of G over a 512-row K-chunk.
// grid = (512/64, 512/32, 16 K-chunks), block = 1 wave32
__global__ void kgram(const float* __restrict__ xn, const float* __restrict__ x,
                      float* __restrict__ G) {
    int lane = threadIdx.x;
    int half = lane >> 4, i16 = lane & 15;
    int nb = blockIdx.x;                 // col block: nb*64
    int mb = blockIdx.y;                 // row block: mb*32
    int kc = blockIdx.z;                 // K chunk:  kc*512
    // A(16x4) fragment: lane -> (M=i16, K=v+2*half); A[M][K] = xn[k][mb*32+16*mt+M]
    // B(4x16) fragment: lane -> (K=v+2*half, N=i16); B[K][N] = x[k][nb*64+16*nt+N]
    const float* pa = xn + (size_t)(kc * 512 + 2 * half) * DIM + mb * 32 + i16;
    const float* pb = x  + (size_t)(kc * 512 + 2 * half) * DIM + nb * 64 + i16;
    v8f acc[2][4] = {};
    for (int k = 0; k < 512; k += 4) {
        v2f a[2], b[4];
#pragma unroll
        for (int mt = 0; mt < 2; ++mt) { a[mt].x = pa[16 * mt]; a[mt].y = pa[16 * mt + DIM]; }
#pragma unroll
        for (int nt = 0; nt < 4; ++nt) { b[nt].x = pb[16 * nt]; b[nt].y = pb[16 * nt + DIM]; }
        pa += 4 * DIM; pb += 4 * DIM;
#pragma unroll
        for (int mt = 0; mt < 2; ++mt)
#pragma unroll
            for (int nt = 0; nt < 4; ++nt)
                acc[mt][nt] = __builtin_amdgcn_wmma_f32_16x16x4_f32(
                    false, a[mt], false, b[nt], (short)0, acc[mt][nt], false, false);
    }
    // C(16x16) f32 layout: VGPR v -> M = v + 8*half, N = i16
#pragma unroll
    for (int mt = 0; mt < 2; ++mt) {
        float* out = G + (size_t)(mb * 32 + 16 * mt + 8 * half) * DIM + nb * 64 + i16;
#pragma unroll
        for (int nt = 0; nt < 4; ++nt)
#pragma unroll
            for (int v = 0; v < 8; ++v)
                atomicAdd(out + (size_t)v * DIM + 16 * nt, acc[mt][nt][v]);
    }
}

// ---------- Kernel 4: x_agg = xn @ G + x, fused weighted class scatter --------
// Each wave computes a 32x64 block of x_agg. grid = (512/64, 8192/32), 1 wave32.
__global__ void kagg(const float* __restrict__ xn, const float* __restrict__ x,
                     const float* __restrict__ G, const float* __restrict__ maxprob,
                     const int* __restrict__ cls, float* __restrict__ sums) {
    int lane = threadIdx.x;
    int half = lane >> 4, i16 = lane & 15;
    int nb = blockIdx.x;                 // col block: nb*64
    int mb = blockIdx.y;                 // row block: mb*32
    // A[M][K] = xn[mb*32+16*mt+M][K]  (x/y contiguous -> b64 loads)
    // B[K][N] = G[K][nb*64+16*nt+N]
    const float* pa = xn + (size_t)(mb * 32 + i16) * DIM + 2 * half;
    const float* pb = G  + (size_t)(2 * half) * DIM + nb * 64 + i16;
    v8f acc[2][4] = {};
    for (int k = 0; k < DIM; k += 4) {
        v2f a[2], b[4];
#pragma unroll
        for (int mt = 0; mt < 2; ++mt) {
            a[mt].x = pa[(size_t)16 * mt * DIM];
            a[mt].y = pa[(size_t)16 * mt * DIM + 1];
        }
#pragma unroll
        for (int nt = 0; nt < 4; ++nt) { b[nt].x = pb[16 * nt]; b[nt].y = pb[16 * nt + DIM]; }
        pa += 4; pb += 4 * DIM;
#pragma unroll
        for (int mt = 0; mt < 2; ++mt)
#pragma unroll
            for (int nt = 0; nt < 4; ++nt)
                acc[mt][nt] = __builtin_amdgcn_wmma_f32_16x16x4_f32(
                    false, a[mt], false, b[nt], (short)0, acc[mt][nt], false, false);
    }
#pragma unroll
    for (int mt = 0; mt < 2; ++mt) {
#pragma unroll
        for (int v = 0; v < 8; ++v) {
            int   row = mb * 32 + 16 * mt + v + 8 * half;
            float p   = maxprob[row];
            int   cl  = cls[row];
#pragma unroll
            for (int nt = 0; nt < 4; ++nt) {
                int   c  = nb * 64 + 16 * nt + i16;
                float xa = acc[mt][nt][v] + x[(size_t)row * DIM + c];   // + x residual
                atomicAdd(&sums[(size_t)cl * DIM + c], xa * p);
            }
        }
    }
}

// ---------- Kernel 5: prototypes = sums / max(counts,1), 0 where empty ----------
__global__ void kproto(const float* __restrict__ sums, const float* __restrict__ counts,
                       float* __restrict__ proto) {
    int idx = blockIdx.x * blockDim.x + threadIdx.x;
    if (idx >= NCLS * DIM) return;
    float cnt = counts[idx / DIM];
    proto[idx] = (cnt > 0.f) ? sums[idx] / fmaxf(cnt, 1.0f) : 0.f;
}

// ---------- Kernel 6: inter[i][j][d] = proto[j][d] - proto[i][d] ----------
__global__ void kinter(const float* __restrict__ proto, float* __restrict__ out) {
    size_t idx = (size_t)blockIdx.x * blockDim.x + threadIdx.x;
    if (idx >= (size_t)NCLS * NCLS * DIM) return;
    int    d  = (int)(idx % DIM);
    size_t ij = idx / DIM;
    int    j  = (int)(ij % NCLS);
    int    i  = (int)(ij / NCLS);
    out[idx] = proto[(size_t)j * DIM + d] - proto[(size_t)i * DIM + d];
}

extern "C" void kernel_launch(void* const* d_in, const int* in_sizes, int n_in,
                              void* d_out, int out_size, void* d_ws, size_t ws_size,
                              hipStream_t stream) {
    const float* x      = (const float*)d_in[0];
    const float* logits = (const float*)d_in[1];
    float* out = (float*)d_out;

    // workspace layout (floats): ~18.1 MB total
    float* ws      = (float*)d_ws;
    float* xn      = ws;                                    // 8192*512
    float* G       = xn + (size_t)N_ROWS * DIM;             // 512*512
    float* sums    = G + (size_t)DIM * DIM;                 // 100*512
    float* counts  = sums + (size_t)NCLS * DIM;             // 100
    float* maxprob = counts + NCLS;                         // 8192
    int*   cls     = (int*)(maxprob + N_ROWS);              // 8192

    // zero the atomic accumulators (G, sums, counts are contiguous)
    hipMemsetAsync(G, 0,
                   ((size_t)DIM * DIM + (size_t)NCLS * DIM + NCLS) * sizeof(float),
                   stream);

    knorm<<<N_ROWS, 32, 0, stream>>>(x, xn);
    ksoft<<<N_ROWS, 32, 0, stream>>>(logits, maxprob, cls, counts);
    kgram<<<dim3(DIM / 64, DIM / 32, 16), 32, 0, stream>>>(xn, x, G);
    kagg<<<dim3(DIM / 64, N_ROWS / 32), 32, 0, stream>>>(xn, x, G, maxprob, cls, sums);

    float* proto = out;                                     // first 100*512 of d_out
    kproto<<<(NCLS * DIM + 255) / 256, 256, 0, stream>>>(sums, counts, proto);
    size_t ninter = (size_t)NCLS * NCLS * DIM;
    kinter<<<(unsigned)((ninter + 255) / 256), 256, 0, stream>>>(proto, out + (size_t)NCLS * DIM);
}